// ModifiedCTCLoss_13280038880001
// MI455X (gfx1250) — compile-verified
//
#include <hip/hip_runtime.h>
#include <hip/hip_bf16.h>
#include <stdint.h>

// Problem constants (match reference)
#define T_DIM 1600
#define N_DIM 32
#define C_DIM 512
#define S_DIM 200
#define L_DIM 401          // 2*S + 1
#define BLOCK 416          // 13 wave32s; one thread per trellis state
#define DEPTH 16           // async prefetch depth (rows of 2 KB -> 32 KB LDS ring)
#define NEGF (-1e30f)
#define LOG2E 1.4426950408889634f
#define LN2   0.6931471805599453f

// CDNA5 async global->LDS copy: 16 bytes per active lane, tracked by ASYNCcnt.
// Non-temporal: log_probs is streamed exactly once (105 MB).
__device__ __forceinline__ void async_load_16B_to_lds(uint32_t lds_off,
                                                      const void* gaddr) {
  asm volatile("global_load_async_to_lds_b128 %0, %1, off th:TH_LOAD_NT"
               :: "v"(lds_off), "v"((unsigned long long)(uintptr_t)gaddr)
               : "memory");
}

// Bare hardware transcendentals: v_exp_f32 (2^x) and v_log_f32 (log2 x),
// no base-conversion multiplies on the serial chain.
__device__ __forceinline__ float exp2_hw(float x) { return __builtin_amdgcn_exp2f(x); }
__device__ __forceinline__ float log2_hw(float x) { return __builtin_amdgcn_logf(x); }

__global__ __launch_bounds__(BLOCK)
void ctc_alpha_kernel(const float* __restrict__ log_probs,   // (T, N, C)
                      const int* __restrict__ targets,       // (N, S)
                      const int* __restrict__ input_lens,    // (N,)
                      const int* __restrict__ target_lens,   // (N,)
                      float* __restrict__ per_ex) {          // (N,)
  __shared__ float rows_s[DEPTH][C_DIM];   // emission-row ring buffer (32 KB)
  __shared__ float beta_s[2][BLOCK];       // double-buffered trellis, log2 domain

  const int n   = blockIdx.x;
  const int tid = threadIdx.x;
  const int s   = tid;
  const bool is_state = (s < L_DIM);

  const int ilen = input_lens[n];
  const int tlen = target_lens[n];

  // Per-thread extended label class + skip (s-2 -> s) legality.
  int  ext_c = 0;          // BLANK for even s
  bool allow_skip = false; // only odd s >= 3 with differing neighbor labels
  if (is_state && (s & 1)) {
    ext_c = targets[n * S_DIM + ((s - 1) >> 1)];
    if (s >= 3) {
      const int prev = targets[n * S_DIM + ((s - 3) >> 1)];
      allow_skip = (ext_c != prev);   // labels are in [1,C): never BLANK
    }
  }

  const float* row_base   = log_probs + (size_t)n * C_DIM;
  const size_t row_stride = (size_t)N_DIM * C_DIM;   // floats between timesteps

  const uint32_t lane_off = (uint32_t)tid * 16u;     // 128 lanes x 16 B = 2 KB row

  // Prologue: prefetch rows 0..DEPTH-1 (DEPTH copies in flight).
  if (tid < 128) {
    for (int d = 0; d < DEPTH; ++d) {
      const uint32_t l = (uint32_t)(uintptr_t)(&rows_s[d][0]) + lane_off;
      const void*    g = (const char*)(row_base + (size_t)d * row_stride) + lane_off;
      async_load_16B_to_lds(l, g);
    }
  }

  // ---- t = 0 (peeled): initialize the trellis into buffer 1 ----
  asm volatile("s_wait_asynccnt 15" ::: "memory");   // row 0 landed
  __syncthreads();
  if (is_state) {
    const float e2 = rows_s[0][ext_c] * LOG2E;       // log2-domain emission
    float newv = NEGF;
    if (s == 0) newv = e2;
    else if (s == 1 && tlen >= 1) newv = e2;
    beta_s[1][s] = newv;
  }
  int cur = 1;

  // ---- steady state: ONE barrier per timestep ----
  // At iteration t we (a) retire row t, (b) re-issue row t+DEPTH-1 into the
  // slot consumed at t-1 (the barrier proves all reads of it are done).
  for (int t = 1; t < T_DIM; ++t) {
    const int slot = t & (DEPTH - 1);

    // Wait for row t. Steady state keeps <=14 outstanding; drain once at tail.
    if (t <= T_DIM - DEPTH + 1) {
      asm volatile("s_wait_asynccnt 14" ::: "memory");
    } else {
      asm volatile("s_wait_asynccnt 0" ::: "memory");
    }
    __syncthreads();   // row t visible; beta[cur] settled; slot(t-1) reusable

    // Refill: row t+DEPTH-1 into the slot consumed last iteration.
    const int tp = t + DEPTH - 1;
    if (tp < T_DIM && tid < 128) {
      const uint32_t l =
          (uint32_t)(uintptr_t)(&rows_s[tp & (DEPTH - 1)][0]) + lane_off;
      const void* g =
          (const char*)(row_base + (size_t)tp * row_stride) + lane_off;
      async_load_16B_to_lds(l, g);
    }

    if (is_state) {
      const float e2 = rows_s[slot][ext_c] * LOG2E;  // off the critical path
      const float b0 = beta_s[cur][s];
      const float b1 = (s >= 1)   ? beta_s[cur][s - 1] : NEGF;
      const float b2 = allow_skip ? beta_s[cur][s - 2] : NEGF;
      const float m   = fmaxf(b0, fmaxf(b1, b2));
      const float lse = m + log2_hw(exp2_hw(b0 - m) + exp2_hw(b1 - m) + exp2_hw(b2 - m));
      const float newv = (t < ilen) ? (lse + e2) : b0;  // freeze past input_len
      beta_s[cur ^ 1][s] = newv;
    }
    cur ^= 1;
  }

  __syncthreads();   // publish final trellis to lane 0
  if (tid == 0) {
    const int il = 2 * tlen;
    const int ip = (il - 1 > 0) ? (il - 1) : 0;
    const float bl = beta_s[cur][il];
    const float bp = (tlen > 0) ? beta_s[cur][ip] : NEGF;
    const float m  = fmaxf(bl, bp);
    const float l2 = m + log2_hw(exp2_hw(bl - m) + exp2_hw(bp - m));
    per_ex[n] = -LN2 * l2;   // back to natural-log NLL
  }
}

// zero_infinity mask, len^0.5 normalization, mean over N (single wave32).
__global__ __launch_bounds__(32)
void ctc_finalize_kernel(const float* __restrict__ per_ex,
                         const int* __restrict__ target_lens,
                         float* __restrict__ out) {
  const int i = threadIdx.x;
  float v = 0.0f;
  if (i < N_DIM) {
    float p = per_ex[i];
    p = (p < 1e29f) ? p : 0.0f;
    const float norm = sqrtf(fmaxf((float)target_lens[i], 1.0f));
    v = p / norm;
  }
  for (int off = 16; off > 0; off >>= 1)
    v += __shfl_xor(v, off, 32);
  if (i == 0) out[0] = v / (float)N_DIM;
}

extern "C" void kernel_launch(void* const* d_in, const int* in_sizes, int n_in,
                              void* d_out, int out_size, void* d_ws, size_t ws_size,
                              hipStream_t stream) {
  (void)in_sizes; (void)n_in; (void)out_size; (void)ws_size;
  const float* log_probs   = (const float*)d_in[0];
  const int*   targets     = (const int*)d_in[1];
  const int*   input_lens  = (const int*)d_in[2];
  const int*   target_lens = (const int*)d_in[3];
  float* per_ex = (float*)d_ws;      // N floats of scratch
  float* out    = (float*)d_out;

  ctc_alpha_kernel<<<N_DIM, BLOCK, 0, stream>>>(log_probs, targets, input_lens,
                                                target_lens, per_ex);
  ctc_finalize_kernel<<<1, 32, 0, stream>>>(per_ex, target_lens, out);
}